// Receiver_30150670418387
// MI455X (gfx1250) — compile-verified
//
#include <hip/hip_runtime.h>
#include <hip/hip_bf16.h>
#include <math.h>
#include <stdint.h>

// Problem constants from the reference: H=2, C=64, HC=128, HID=256, B=1024.
#define HC   128
#define HID_ 256
#define NB   1024   // batch B (softmax width)

typedef __attribute__((ext_vector_type(2))) float v2f;
typedef __attribute__((ext_vector_type(8))) float v8f;

// ---------------------------------------------------------------------------
// float atomic max via monotonic integer encoding (always compiles)
// ---------------------------------------------------------------------------
__device__ __forceinline__ void atomicMaxFloat(float* addr, float v) {
    if (v >= 0.0f) {
        atomicMax((int*)addr, __float_as_int(v));
    } else {
        atomicMin((unsigned int*)addr, __float_as_uint(v));
    }
}

// ---------------------------------------------------------------------------
// init per-node accumulators: lmax=-inf, denom=0, tsum=0   (shape [N,2] each)
// ---------------------------------------------------------------------------
__global__ void init_node_acc_kernel(float* __restrict__ lmax,
                                     float* __restrict__ denom,
                                     float* __restrict__ tsum, int n2) {
    int i = blockIdx.x * blockDim.x + threadIdx.x;
    if (i < n2) {
        lmax[i]  = -INFINITY;
        denom[i] = 0.0f;
        tsum[i]  = 0.0f;
    }
}

// ---------------------------------------------------------------------------
// per-edge logit computation (shared helper)
// ---------------------------------------------------------------------------
__device__ __forceinline__ void edge_logits(float xs, float xd, float ea,
                                            const float* sWl, const float* sbl,
                                            const float* sWr, const float* sbr,
                                            const float* sWe, const float* satt,
                                            float& l0, float& l1) {
    l0 = 0.0f; l1 = 0.0f;
    #pragma unroll 8
    for (int c = 0; c < 64; ++c) {
        float m0 = xs * sWl[c] + sbl[c] + xd * sWr[c] + sbr[c] + ea * sWe[c];
        m0 = m0 > 0.0f ? m0 : 0.2f * m0;
        l0 += m0 * satt[c];
        float m1 = xs * sWl[64 + c] + sbl[64 + c] + xd * sWr[64 + c] + sbr[64 + c] + ea * sWe[64 + c];
        m1 = m1 > 0.0f ? m1 : 0.2f * m1;
        l1 += m1 * satt[64 + c];
    }
}

// ---------------------------------------------------------------------------
// pass 1 over edges: segment max of logits per dst node
// ---------------------------------------------------------------------------
__global__ __launch_bounds__(256) void edge_max_kernel(
    const float* __restrict__ x, const int* __restrict__ edge_index,
    const float* __restrict__ edge_attr,
    const float* __restrict__ Wl, const float* __restrict__ bl,
    const float* __restrict__ Wr, const float* __restrict__ br,
    const float* __restrict__ We, const float* __restrict__ att,
    float* __restrict__ lmax, int E) {
    __shared__ float sWl[HC], sbl[HC], sWr[HC], sbr[HC], sWe[HC], satt[HC];
    for (int i = threadIdx.x; i < HC; i += blockDim.x) {
        sWl[i] = Wl[i]; sbl[i] = bl[i]; sWr[i] = Wr[i];
        sbr[i] = br[i]; sWe[i] = We[i]; satt[i] = att[i];
    }
    __syncthreads();
    int e = blockIdx.x * blockDim.x + threadIdx.x;
    if (e >= E) return;
    int s = edge_index[e];
    int d = edge_index[E + e];
    float xs = x[s], xd = x[d], ea = edge_attr[e];
    float l0, l1;
    edge_logits(xs, xd, ea, sWl, sbl, sWr, sbr, sWe, satt, l0, l1);
    atomicMaxFloat(&lmax[d * 2 + 0], l0);
    atomicMaxFloat(&lmax[d * 2 + 1], l1);
}

// ---------------------------------------------------------------------------
// pass 2 over edges: ex = exp(logit - lmax[dst]);
//   denom[dst,h] += ex ;  tsum[dst,h] += ex * x[src]
// ---------------------------------------------------------------------------
__global__ __launch_bounds__(256) void edge_sum_kernel(
    const float* __restrict__ x, const int* __restrict__ edge_index,
    const float* __restrict__ edge_attr,
    const float* __restrict__ Wl, const float* __restrict__ bl,
    const float* __restrict__ Wr, const float* __restrict__ br,
    const float* __restrict__ We, const float* __restrict__ att,
    const float* __restrict__ lmax,
    float* __restrict__ denom, float* __restrict__ tsum, int E) {
    __shared__ float sWl[HC], sbl[HC], sWr[HC], sbr[HC], sWe[HC], satt[HC];
    for (int i = threadIdx.x; i < HC; i += blockDim.x) {
        sWl[i] = Wl[i]; sbl[i] = bl[i]; sWr[i] = Wr[i];
        sbr[i] = br[i]; sWe[i] = We[i]; satt[i] = att[i];
    }
    __syncthreads();
    int e = blockIdx.x * blockDim.x + threadIdx.x;
    if (e >= E) return;
    int s = edge_index[e];
    int d = edge_index[E + e];
    float xs = x[s], xd = x[d], ea = edge_attr[e];
    float l0, l1;
    edge_logits(xs, xd, ea, sWl, sbl, sWr, sbr, sWe, satt, l0, l1);
    float ex0 = __expf(l0 - lmax[d * 2 + 0]);
    float ex1 = __expf(l1 - lmax[d * 2 + 1]);
    atomicAdd(&denom[d * 2 + 0], ex0);
    atomicAdd(&tsum[d * 2 + 0], ex0 * xs);
    atomicAdd(&denom[d * 2 + 1], ex1);
    atomicAdd(&tsum[d * 2 + 1], ex1 * xs);
}

// ---------------------------------------------------------------------------
// rebuild g[n, hc] = Wl[hc]*sx + bl[hc]*s1 + bias_out[hc]
//   sx = tsum/(denom+1e-16), s1 = denom/(denom+1e-16)
// ---------------------------------------------------------------------------
__global__ void build_g_kernel(const float* __restrict__ Wl,
                               const float* __restrict__ bl,
                               const float* __restrict__ bias_out,
                               const float* __restrict__ denom,
                               const float* __restrict__ tsum,
                               float* __restrict__ g, int N) {
    long idx = (long)blockIdx.x * blockDim.x + threadIdx.x;
    if (idx >= (long)N * HC) return;
    int n = (int)(idx >> 7);
    int hc = (int)(idx & (HC - 1));
    int h = hc >> 6;
    float den = denom[n * 2 + h];
    float inv = 1.0f / (den + 1e-16f);
    float sx = tsum[n * 2 + h] * inv;
    float s1 = den * inv;
    g[idx] = Wl[hc] * sx + bl[hc] * s1 + bias_out[hc];
}

// ---------------------------------------------------------------------------
// me = message @ fc_W.T + fc_b  :  (1024 x 256) @ (256 x 128) -> meT[128,1024]
// One 16x16 output tile per wave, V_WMMA_F32_16X16X4_F32, K loop of 64.
// ---------------------------------------------------------------------------
__global__ __launch_bounds__(256) void me_wmma_kernel(
    const float* __restrict__ message,  // [1024, 256]
    const float* __restrict__ fcW,      // [128, 256] row-major (j, k)
    const float* __restrict__ fcb,      // [128]
    float* __restrict__ meT) {          // [128, 1024]  (k-major for phase C)
    const int tid  = threadIdx.x;
    const int lane = tid & 31;
    const int ln   = lane & 15;
    const int hi   = lane >> 4;     // lane half selects K pair
    const int w    = blockIdx.x * 8 + (tid >> 5);   // 0..511
    const int tile_b = (w & 63) * 16;               // row tile (batch dim)
    const int tile_j = (w >> 6) * 16;               // col tile (HC dim)

    v8f acc = {};
    const float* ap = message + (size_t)(tile_b + ln) * HID_;
    const float* bp = fcW + (size_t)(tile_j + ln) * HID_;   // B[k,j] = fcW[j,k]
    for (int k = 0; k < HID_; k += 4) {
        v2f a, b;
        a.x = ap[k + 2 * hi];
        a.y = ap[k + 2 * hi + 1];
        b.x = bp[k + 2 * hi];
        b.y = bp[k + 2 * hi + 1];
        acc = __builtin_amdgcn_wmma_f32_16x16x4_f32(false, a, false, b,
                                                    (short)0, acc, false, false);
    }
    const int j = tile_j + ln;
    const float bias = fcb[j];
    #pragma unroll
    for (int v = 0; v < 8; ++v) {
        int bb = tile_b + v + (hi ? 8 : 0);         // C layout: VGPR v -> row m
        meT[j * NB + bb] = acc[v] + bias;           // store transposed
    }
}

// ---------------------------------------------------------------------------
// dots = g @ meT (K=128) for a 16-row strip, then in-LDS row softmax, stream
// out once (output write 205MB dominates -> bandwidth bound).
// The 8KB g strip is staged once per block via the CDNA5 async global->LDS
// path (GLOBAL_LOAD_ASYNC_TO_LDS_B128, ASYNCcnt) instead of being re-read by
// all 8 waves; A fragments then come from LDS (ds_load_b64).
// 8 waves/block; wave w owns columns [w*128, w*128+128) = 8 WMMA tiles.
// ---------------------------------------------------------------------------
__global__ __launch_bounds__(256) void dots_softmax_kernel(
    const float* __restrict__ g,    // [N, 128]
    const float* __restrict__ meT,  // [128, 1024]
    float* __restrict__ out,        // [N, 1024]
    int N) {
    __shared__ float sg[16 * HC];       // 8 KB staged A strip
    __shared__ float sdots[16 * NB];    // 64 KB result strip
    __shared__ float srowmax[16];
    __shared__ float srowsum[16];

    const int tid  = threadIdx.x;
    const int wave = tid >> 5;
    const int lane = tid & 31;
    const int ln   = lane & 15;
    const int hi   = lane >> 4;
    const int row0 = blockIdx.x * 16;
    const int colbase = wave * 128;

    // ---- stage the 16x128 g strip into LDS ----
    if (row0 + 16 <= N) {
        // async copy: 256 threads x 2 x B128 = 8192 bytes
        const char* gsrc = (const char*)(g + (size_t)row0 * HC);
        uint32_t l0 = (uint32_t)(uintptr_t)(void*)&sg[0] + (uint32_t)(tid * 16);
        uint32_t l1 = l0 + 4096u;
        uint64_t a0 = (uint64_t)(uintptr_t)(gsrc + tid * 16);
        uint64_t a1 = a0 + 4096u;
        asm volatile("global_load_async_to_lds_b128 %0, %1, off"
                     :: "v"(l0), "v"(a0) : "memory");
        asm volatile("global_load_async_to_lds_b128 %0, %1, off"
                     :: "v"(l1), "v"(a1) : "memory");
        asm volatile("s_wait_asynccnt 0x0" ::: "memory");
    } else {
        // ragged tail fallback: guarded scalar staging
        for (int i = tid; i < 16 * HC; i += 256) {
            int r = row0 + (i >> 7);
            if (r >= N) r = N - 1;
            sg[i] = g[(size_t)r * HC + (i & (HC - 1))];
        }
    }
    __syncthreads();

    v8f acc[8] = {};
    const float* aptr = sg + ln * HC;   // A row for this lane (from LDS)

    for (int k = 0; k < HC; k += 4) {
        v2f a;
        a.x = aptr[k + 2 * hi];
        a.y = aptr[k + 2 * hi + 1];
        const float* b0 = meT + (size_t)(k + 2 * hi) * NB;
        #pragma unroll
        for (int t = 0; t < 8; ++t) {
            int col = colbase + t * 16 + ln;
            v2f b;
            b.x = b0[col];
            b.y = b0[NB + col];
            acc[t] = __builtin_amdgcn_wmma_f32_16x16x4_f32(false, a, false, b,
                                                           (short)0, acc[t],
                                                           false, false);
        }
    }

    // spill C fragments to the LDS strip
    #pragma unroll
    for (int t = 0; t < 8; ++t) {
        int col = colbase + t * 16 + ln;
        #pragma unroll
        for (int v = 0; v < 8; ++v) {
            int m = v + (hi ? 8 : 0);
            sdots[m * NB + col] = acc[t][v];
        }
    }
    __syncthreads();

    // row max: wave w reduces rows 2w, 2w+1
    #pragma unroll
    for (int rr = 0; rr < 2; ++rr) {
        int r = wave * 2 + rr;
        float mx = -INFINITY;
        for (int c = lane; c < NB; c += 32) mx = fmaxf(mx, sdots[r * NB + c]);
        for (int off = 16; off > 0; off >>= 1) mx = fmaxf(mx, __shfl_xor(mx, off, 32));
        if (lane == 0) srowmax[r] = mx;
    }
    __syncthreads();

    // exp + row sum (store exp back into LDS)
    #pragma unroll
    for (int rr = 0; rr < 2; ++rr) {
        int r = wave * 2 + rr;
        float mx = srowmax[r];
        float s = 0.0f;
        for (int c = lane; c < NB; c += 32) {
            float ev = __expf(sdots[r * NB + c] - mx);
            sdots[r * NB + c] = ev;
            s += ev;
        }
        for (int off = 16; off > 0; off >>= 1) s += __shfl_xor(s, off, 32);
        if (lane == 0) srowsum[r] = s;
    }
    __syncthreads();

    // normalized stream-out, coalesced
    for (int idx = tid; idx < 16 * NB; idx += 256) {
        int r = idx >> 10;
        int c = idx & (NB - 1);
        int row = row0 + r;
        if (row < N) out[(size_t)row * NB + c] = sdots[idx] / srowsum[r];
    }
}

// ---------------------------------------------------------------------------
extern "C" void kernel_launch(void* const* d_in, const int* in_sizes, int n_in,
                              void* d_out, int out_size, void* d_ws, size_t ws_size,
                              hipStream_t stream) {
    const float* x         = (const float*)d_in[0];
    const int*   edge_idx  = (const int*)d_in[1];
    const float* edge_attr = (const float*)d_in[2];
    const float* message   = (const float*)d_in[3];
    const float* Wl        = (const float*)d_in[4];
    const float* bl        = (const float*)d_in[5];
    const float* Wr        = (const float*)d_in[6];
    const float* br        = (const float*)d_in[7];
    const float* We        = (const float*)d_in[8];
    const float* att       = (const float*)d_in[9];
    const float* bias_out  = (const float*)d_in[10];
    const float* fcW       = (const float*)d_in[11];
    const float* fcb       = (const float*)d_in[12];

    const int N = in_sizes[0];        // x is (N,1)
    const int E = in_sizes[1] / 2;    // edge_index is (2,E)

    // workspace layout
    float* lmax  = (float*)d_ws;                 // [N,2]
    float* denom = lmax  + (size_t)N * 2;        // [N,2]
    float* tsum  = denom + (size_t)N * 2;        // [N,2]
    float* meT   = tsum  + (size_t)N * 2;        // [128,1024]
    float* g     = meT   + (size_t)HC * NB;      // [N,128]

    float* out = (float*)d_out;

    int n2 = N * 2;
    init_node_acc_kernel<<<(n2 + 255) / 256, 256, 0, stream>>>(lmax, denom, tsum, n2);

    int eblocks = (E + 255) / 256;
    edge_max_kernel<<<eblocks, 256, 0, stream>>>(x, edge_idx, edge_attr,
                                                 Wl, bl, Wr, br, We, att, lmax, E);
    edge_sum_kernel<<<eblocks, 256, 0, stream>>>(x, edge_idx, edge_attr,
                                                 Wl, bl, Wr, br, We, att,
                                                 lmax, denom, tsum, E);

    long gtot = (long)N * HC;
    build_g_kernel<<<(int)((gtot + 255) / 256), 256, 0, stream>>>(Wl, bl, bias_out,
                                                                  denom, tsum, g, N);

    me_wmma_kernel<<<64, 256, 0, stream>>>(message, fcW, fcb, meT);

    dots_softmax_kernel<<<(N + 15) / 16, 256, 0, stream>>>(g, meT, out, N);
}